// PcPreprocessor3D_13417477833539
// MI455X (gfx1250) — compile-verified
//
#include <hip/hip_runtime.h>
#include <hip/hip_bf16.h>

// ---------------------------------------------------------------------------
// PcPreprocessor3D for MI455X (gfx1250).
// Memory-bound streaming kernel: 8M points, ~0.96 GB total traffic -> ~41us
// at 23.3 TB/s. No matrix math, so no WMMA; we use the CDNA5 data-movement
// paths instead: TDM (tensor_load_to_lds) double-buffered LDS staging with
// s_wait_tensorcnt, and non-temporal (NT) cache hints on the streamed output.
// Toolchain probe (round 1): 6-arg tensor_load_to_lds -> amdgpu-toolchain.
// ---------------------------------------------------------------------------

typedef float v4f __attribute__((ext_vector_type(4)));
typedef float v2f __attribute__((ext_vector_type(2)));

#ifndef __has_builtin
#define __has_builtin(x) 0
#endif

#if __has_builtin(__builtin_amdgcn_tensor_load_to_lds) && \
    __has_builtin(__builtin_amdgcn_s_wait_tensorcnt)
#define USE_TDM 1
#else
#define USE_TDM 0
#endif

// Per-point compute + output emission.
// Output layout (flat float elements, P = B*N points):
//   [0, 4P)                 pc passthrough (float4 per point)
//   per scale s in {0.5, 1.0}, base = 4P + s*11P:
//     base + p              top_idx     (int value as float, exact: max 2^24-1)
//     base +  P + 2p        top_coord   (xi, yi)
//     base + 3P + 2p        top_res     (xr, yr)
//     base + 5P + p         front_idx
//     base + 6P + 3p        front_coord (xi, yi, zi)
//     base + 9P + 2p        front_res   (xr, zr)
__device__ __forceinline__ void emit_point(const v4f v, int p,
                                           float* __restrict__ out, int P) {
  // pc copy (b128 NT store, 16B aligned)
  __builtin_nontemporal_store(v, reinterpret_cast<v4f*>(out) + p);

  int base = 4 * P;
#pragma unroll
  for (int s = 0; s < 2; ++s) {
    // scale 0.5 -> sizes (512, 512, 64); scale 1.0 -> (256, 256, 32)
    const float fsx = (s == 0) ? 512.0f : 256.0f;   // sy == sx numerically
    const float fsz = (s == 0) ? 64.0f  : 32.0f;
    const int   isx = (s == 0) ? 512 : 256;
    const int   isz = (s == 0) ? 64  : 32;

    // Mirror reference: idx = (x - min) / (max - min) * size; trunc to int.
    float qx = (v.x - 0.0f)  / 51.2f * fsx;
    float qy = (v.y + 25.6f) / 51.2f * fsx;
    float qz = (v.z + 2.0f)  / 6.4f  * fsz;
    int   xi = (int)qx, yi = (int)qy, zi = (int)qz;
    float xr = qx - (float)xi;
    float yr = qy - (float)yi;
    float zr = qz - (float)zi;

    int top   = xi * isx + yi;           // uses X size for Y stride (faithful)
    int front = top * isz + zi;          // == xi*sx*sz + yi*sz + zi

    float* o = out + base;
    __builtin_nontemporal_store((float)top, o + p);

    v2f tc; tc[0] = (float)xi; tc[1] = (float)yi;
    __builtin_nontemporal_store(tc, reinterpret_cast<v2f*>(o + (size_t)P) + p);

    v2f tr; tr[0] = xr; tr[1] = yr;
    __builtin_nontemporal_store(tr, reinterpret_cast<v2f*>(o + (size_t)3 * P) + p);

    __builtin_nontemporal_store((float)front, o + (size_t)5 * P + p);

    // stride-3: three b32 NT stores (vec3 would pad to 16B and break layout)
    float* fc = o + (size_t)6 * P + (size_t)3 * p;
    __builtin_nontemporal_store((float)xi, fc);
    __builtin_nontemporal_store((float)yi, fc + 1);
    __builtin_nontemporal_store((float)zi, fc + 2);

    v2f fr; fr[0] = xr; fr[1] = zr;
    __builtin_nontemporal_store(fr, reinterpret_cast<v2f*>(o + (size_t)9 * P) + p);

    base += 11 * P;
  }
}

#if USE_TDM
typedef unsigned int u32x4 __attribute__((ext_vector_type(4)));
typedef int          i32x8 __attribute__((ext_vector_type(8)));
typedef int          i32x4 __attribute__((ext_vector_type(4)));

// Issue one TDM load: a 1-D tile of 128 points (2048 B = 256 x 8B units)
// from global memory into LDS at byte offset lds_byte_off.
__device__ __forceinline__ void tdm_tile_load(const v4f* gsrc,
                                              unsigned lds_byte_off) {
  unsigned long long ga = (unsigned long long)(const void*)gsrc;
  const unsigned units = 256u;                      // 8-byte units per tile

  // D# group 0 (128b): count=1 | lds_addr | global_addr[56:0] | type=2
  u32x4 g0;
  g0[0] = 1u;                                       // count=1, user descriptor
  g0[1] = lds_byte_off;                             // lds_addr (bytes)
  g0[2] = (unsigned)ga;                             // global_addr[31:0]
  g0[3] = (unsigned)((ga >> 32) & 0x01FFFFFFu)      // global_addr[56:32]
        | (2u << 30);                               // type=2 ("image")

  // D# group 1 (256b): data_size=3 (8B); 1-D tile of `units`; stride=units
  i32x8 g1;
  g1[0] = (int)(3u << 16);                          // data_size=3, no wg mask
  g1[1] = (int)((units & 0xFFFFu) << 16);           // tensor_dim0[15:0]
  g1[2] = (int)(((units >> 16) & 0xFFFFu)           // tensor_dim0[31:16]
        | (1u << 16));                              // tensor_dim1 = 1
  g1[3] = (int)(units << 16);                       // tile_dim0 = units
  g1[4] = 0;                                        // tile_dim1/2 unused (1-D)
  g1[5] = (int)units;                               // tensor_dim0_stride
  g1[6] = 0;
  g1[7] = 0;

  i32x4 g2 = {0, 0, 0, 0};                          // groups 2/3 unused (<=2D)
  i32x4 g3 = {0, 0, 0, 0};
  i32x8 g4 = {0, 0, 0, 0, 0, 0, 0, 0};              // extra group (clang-23 form)
  __builtin_amdgcn_tensor_load_to_lds(g0, g1, g2, g3, g4, 0);
}

__global__ __launch_bounds__(256) void pc_prep_tdm(const v4f* __restrict__ pc,
                                                   float* __restrict__ out,
                                                   int P) {
  // 8 waves * 2 buffers * 128 points * 16B = 32 KB LDS
  __shared__ v4f smem[2048];

  const int lane = threadIdx.x & 31;
  const int wv   = threadIdx.x >> 5;
  const int gw   = blockIdx.x * 8 + wv;        // global wave id
  const int nW   = gridDim.x * 8;
  const int nTiles = P >> 7;                   // 128-point tiles

  v4f* wbuf = &smem[wv * 256];                 // this wave's two buffers
  unsigned lds0 = (unsigned)(unsigned long long)(void*)&wbuf[0];
  unsigned lds1 = (unsigned)(unsigned long long)(void*)&wbuf[128];

  if (gw < nTiles) tdm_tile_load(pc + (size_t)gw * 128, lds0);
  int parity = 0;

  for (int t = gw; t < nTiles; t += nW) {
    int tn = t + nW;
    if (tn < nTiles) {
      // prefetch next tile into the other buffer, then wait for current
      tdm_tile_load(pc + (size_t)tn * 128, parity ? lds0 : lds1);
      __builtin_amdgcn_s_wait_tensorcnt(1);
    } else {
      __builtin_amdgcn_s_wait_tensorcnt(0);
    }
    asm volatile("" ::: "memory");             // keep LDS reads behind the wait

    const v4f* buf = parity ? (wbuf + 128) : wbuf;
    int pbase = t * 128;
#pragma unroll
    for (int k = 0; k < 4; ++k) {
      int j = lane + 32 * k;                   // ds_read_b128 per lane
      v4f v = buf[j];
      emit_point(v, pbase + j, out, P);
    }
    parity ^= 1;
    asm volatile("" ::: "memory");             // don't hoist next issue above reads
  }

  // tail: points beyond the last full 128-point tile
  for (int p = (nTiles << 7) + blockIdx.x * 256 + threadIdx.x; p < P;
       p += gridDim.x * 256) {
    v4f v = __builtin_nontemporal_load(pc + p);
    emit_point(v, p, out, P);
  }
}
#else
__global__ __launch_bounds__(256) void pc_prep_direct(
    const v4f* __restrict__ pc, float* __restrict__ out, int P) {
  int p = blockIdx.x * 256 + threadIdx.x;
  if (p >= P) return;
  if (p + 2048 < P)
    __builtin_prefetch((const void*)(pc + p + 2048), 0, 0);  // global_prefetch_b8
  v4f v = __builtin_nontemporal_load(pc + p);
  emit_point(v, p, out, P);
}
#endif

extern "C" void kernel_launch(void* const* d_in, const int* in_sizes, int n_in,
                              void* d_out, int out_size, void* d_ws, size_t ws_size,
                              hipStream_t stream) {
  (void)n_in; (void)out_size; (void)d_ws; (void)ws_size;
  const v4f* pc = (const v4f*)d_in[0];
  float* out = (float*)d_out;
  const int P = in_sizes[0] / 4;   // B*N points

#if USE_TDM
  // Fixed grid; tiles are grid-strided across 8 waves/block.
  // 1024 blocks = 8192 waves: plenty to saturate HBM, small launch overhead.
  pc_prep_tdm<<<1024, 256, 0, stream>>>(pc, out, P);
#else
  int blocks = (P + 255) / 256;
  pc_prep_direct<<<blocks, 256, 0, stream>>>(pc, out, P);
#endif
}